// CARES_63239098466719
// MI455X (gfx1250) — compile-verified
//
#include <hip/hip_runtime.h>
#include <hip/hip_bf16.h>
#include <math.h>

// CARES forward for MI455X (gfx1250). fp32 WMMA (V_WMMA_F32_16X16X4_F32) for all GEMMs.
#define BB 512
#define LL 50
#define HH 256
#define NNODES (BB*LL)      // 25600 graph nodes
#define NEDGE  (NNODES*12)  // 307200 edges

typedef float v2f __attribute__((ext_vector_type(2)));
typedef float v8f __attribute__((ext_vector_type(8)));

__device__ __forceinline__ unsigned enc_f(float f) {
  unsigned u = __float_as_uint(f);
  return (u & 0x80000000u) ? ~u : (u ^ 0x80000000u);
}
__device__ __forceinline__ float dec_f(unsigned u) {
  return __uint_as_float((u & 0x80000000u) ? (u ^ 0x80000000u) : ~u);
}

// ---------------------------------------------------------------------------
// 1) gather + pooling + alias permute + pos/len encoding
// ---------------------------------------------------------------------------
__global__ void k_prep(const int* __restrict__ alias_in, const int* __restrict__ items,
                       const int* __restrict__ mask, const int* __restrict__ cates,
                       const float* __restrict__ item_emb, const float* __restrict__ cate_emb,
                       const float* __restrict__ pos_emb, const float* __restrict__ len_emb,
                       float* __restrict__ seqb, float* __restrict__ mi,
                       float* __restrict__ mc, float* __restrict__ htl) {
  int b = blockIdx.x, hh = threadIdx.x;
  float ai = 0.f, ac = 0.f, gs = 0.f; int len = 0;
  for (int l = 0; l < LL; ++l) {
    int it = items[b*LL + l], ct = cates[b*LL + l];
    float gm = it > 0 ? 1.f : (it < 0 ? -1.f : 0.f);
    ai += item_emb[(size_t)it*HH + hh] * gm;
    ac += cate_emb[(size_t)ct*HH + hh] * gm;
    gs += gm; len += mask[b*LL + l];
  }
  mi[b*HH + hh] = ai / gs;
  mc[b*HH + hh] = ac / gs;
  const float* le = len_emb + (size_t)len*HH;
  for (int l = 0; l < LL; ++l) {
    int al = alias_in[b*LL + l];
    int it = items[b*LL + al];
    int mk = mask[b*LL + l];
    int p = (len - 1 - l) * mk; if (p < 0) p = 0;
    float v = item_emb[(size_t)it*HH + hh] + pos_emb[p*HH + hh] + le[hh];
    seqb[((size_t)b*LL + l)*HH + hh] = v;
    if (l == len - 1) htl[b*HH + hh] = v;
  }
}

// ---------------------------------------------------------------------------
// 2) zero scratch for GAT scatter
// ---------------------------------------------------------------------------
__global__ void k_zero(float* __restrict__ gout, float* __restrict__ den,
                       unsigned* __restrict__ mx) {
  size_t i = (size_t)blockIdx.x * 256 + threadIdx.x;
  if (i < (size_t)NNODES * HH) gout[i] = 0.f;
  if (i < NNODES) { den[i] = 0.f; mx[i] = 0u; }
}

// ---------------------------------------------------------------------------
// 3) generic WMMA f32 GEMM:  C[rows,256] = gather(A) @ W[K,256] (+bias)
//    one wave per 16x16 output tile; A-row gather via gidx (nullable).
// ---------------------------------------------------------------------------
__global__ void k_gemm(const float* __restrict__ Abase, const int* __restrict__ gidx,
                       const float* __restrict__ W, const float* __restrict__ bias,
                       float* __restrict__ C, int K) {
  int lane = threadIdx.x;
  int half = lane >> 4, lm = lane & 15;
  int r0 = blockIdx.x * 16;
  int n0 = blockIdx.y * 16;
  int arow = r0 + lm;
  int grow = gidx ? gidx[arow] : arow;
  const float* A = Abase + (size_t)grow * K;
  v8f acc = {0.f,0.f,0.f,0.f,0.f,0.f,0.f,0.f};
  for (int kk = 0; kk < K; kk += 4) {
    int ka = kk + 2*half;
    v2f a; a.x = A[ka]; a.y = A[ka+1];
    v2f bfr; bfr.x = W[(size_t)ka*HH + n0 + lm]; bfr.y = W[(size_t)(ka+1)*HH + n0 + lm];
    acc = __builtin_amdgcn_wmma_f32_16x16x4_f32(false, a, false, bfr, (short)0, acc,
                                                false, false);
  }
  float bv = bias ? bias[n0 + lm] : 0.f;
#pragma unroll
  for (int i = 0; i < 8; ++i) {
    int m = i + 8*half;
    C[(size_t)(r0 + m)*HH + n0 + lm] = acc[i] + bv;
  }
}

// ---------------------------------------------------------------------------
// 4) edge logits + segment max (ordered-uint atomicMax). 1 wave per edge.
// ---------------------------------------------------------------------------
__global__ void k_edge_logit(const float* __restrict__ h, const int* __restrict__ eidx,
                             const int* __restrict__ etype, const float* __restrict__ eattr,
                             const float* __restrict__ rel_emb,
                             float* __restrict__ logit, unsigned* __restrict__ mx) {
  int wid = threadIdx.x >> 5, lane = threadIdx.x & 31;
  int e = blockIdx.x * 8 + wid;
  if (e >= NEDGE) return;
  int src = eidx[e], dst = eidx[NEDGE + e];
  const float* hs = h + (size_t)src*HH;
  const float* hd = h + (size_t)dst*HH;
  const float* re = rel_emb + (size_t)etype[e]*HH;
  float dot = 0.f;
  for (int j = lane; j < HH; j += 32) dot += hd[j] * (hs[j] + re[j]);
  for (int off = 16; off >= 1; off >>= 1) dot += __shfl_xor(dot, off, 32);
  if (lane == 0) {
    float x = dot * 0.0625f;                  // 1/sqrt(256)
    float lg = (x > 0.f ? x : 0.2f*x) + eattr[e];
    logit[e] = lg;
    atomicMax(&mx[dst], enc_f(lg));
  }
}

// ---------------------------------------------------------------------------
// 5) exp + segment sum
// ---------------------------------------------------------------------------
__global__ void k_edge_exp(const int* __restrict__ eidx, const float* __restrict__ logit,
                           const unsigned* __restrict__ mx, float* __restrict__ exb,
                           float* __restrict__ den) {
  int e = blockIdx.x * 256 + threadIdx.x;
  if (e >= NEDGE) return;
  int dst = eidx[NEDGE + e];
  float ex = expf(logit[e] - dec_f(mx[dst]));
  exb[e] = ex;
  atomicAdd(&den[dst], ex);
}

// ---------------------------------------------------------------------------
// 6) att * msg scatter-add. 1 block (256 thr) per edge.
// ---------------------------------------------------------------------------
__global__ void k_edge_scatter(const float* __restrict__ h, const int* __restrict__ eidx,
                               const int* __restrict__ etype, const float* __restrict__ exb,
                               const float* __restrict__ den, const float* __restrict__ rel_emb,
                               float* __restrict__ gout) {
  int e = blockIdx.x, j = threadIdx.x;
  int src = eidx[e], dst = eidx[NEDGE + e];
  float att = exb[e] / (den[dst] + 1e-12f);
  float m = h[(size_t)src*HH + j] + rel_emb[(size_t)etype[e]*HH + j];
  atomicAdd(&gout[(size_t)dst*HH + j], att * m);
}

// ---------------------------------------------------------------------------
// 7) graph pooling + alias gather of graph_emb
// ---------------------------------------------------------------------------
__global__ void k_gpool(const int* __restrict__ alias_in, const int* __restrict__ items,
                        const int* __restrict__ mask, const float* __restrict__ gout,
                        float* __restrict__ sn, float* __restrict__ gseq,
                        float* __restrict__ htg) {
  int b = blockIdx.x, hh = threadIdx.x;
  float gs = 0.f, acc = 0.f; int len = 0;
  for (int l = 0; l < LL; ++l) {
    int it = items[b*LL + l];
    float gm = it > 0 ? 1.f : (it < 0 ? -1.f : 0.f);
    gs += gm; len += mask[b*LL + l];
    acc += gout[((size_t)b*LL + l)*HH + hh] * gm;
  }
  sn[b*HH + hh] = acc / gs;
  for (int l = 0; l < LL; ++l) {
    int al = alias_in[b*LL + l];
    float v = gout[((size_t)b*LL + al)*HH + hh];
    gseq[((size_t)b*LL + l)*HH + hh] = v;
    if (l == len - 1) htg[b*HH + hh] = v;
  }
}

// ---------------------------------------------------------------------------
// 8) qc = ht@W0 + mi@W1 + mcx@W3 + b0+b1+b2+b3   (row-constant part of alpha)
// ---------------------------------------------------------------------------
__global__ void k_qc(const float* __restrict__ ht, const float* __restrict__ mi,
                     const float* __restrict__ mcx, const float* __restrict__ Wb,
                     const float* __restrict__ bb, float* __restrict__ qc) {
  int b = blockIdx.x, hh = threadIdx.x;
  __shared__ float sht[HH], smi[HH], smx[HH];
  sht[hh] = ht[b*HH + hh]; smi[hh] = mi[b*HH + hh]; smx[hh] = mcx[b*HH + hh];
  __syncthreads();
  const float* W0 = Wb;
  const float* W1 = Wb + HH*HH;
  const float* W3 = Wb + 3*HH*HH;
  float acc = bb[hh] + bb[HH + hh] + bb[2*HH + hh] + bb[3*HH + hh];
  for (int k = 0; k < HH; ++k)
    acc += sht[k]*W0[k*HH + hh] + smi[k]*W1[k*HH + hh] + smx[k]*W3[k*HH + hh];
  qc[b*HH + hh] = acc;
}

// ---------------------------------------------------------------------------
// 9) fused WMMA attention: alpha[row] = sum_n sigmoid(seq@Ws2 + qc)[n] * w5[n]
//    one wave per 16 rows; loops 16 col-tiles, reduces via 16-lane shuffles.
// ---------------------------------------------------------------------------
__global__ void k_alpha(const float* __restrict__ seqb, const float* __restrict__ qc,
                        const float* __restrict__ W2, const float* __restrict__ w5,
                        float* __restrict__ alpha) {
  int lane = threadIdx.x;
  int half = lane >> 4, lm = lane & 15;
  int r0 = blockIdx.x * 16;
  const float* A = seqb + (size_t)(r0 + lm) * HH;
  float asum[8] = {0.f,0.f,0.f,0.f,0.f,0.f,0.f,0.f};
  for (int t = 0; t < 16; ++t) {
    int n0 = t * 16;
    v8f acc = {0.f,0.f,0.f,0.f,0.f,0.f,0.f,0.f};
    for (int kk = 0; kk < HH; kk += 4) {
      int ka = kk + 2*half;
      v2f a; a.x = A[ka]; a.y = A[ka+1];
      v2f bfr; bfr.x = W2[(size_t)ka*HH + n0 + lm]; bfr.y = W2[(size_t)(ka+1)*HH + n0 + lm];
      acc = __builtin_amdgcn_wmma_f32_16x16x4_f32(false, a, false, bfr, (short)0, acc,
                                                  false, false);
    }
    float wv = w5[n0 + lm];
#pragma unroll
    for (int i = 0; i < 8; ++i) {
      int row = r0 + i + 8*half;
      int bidx = row / LL;
      float q = acc[i] + qc[bidx*HH + n0 + lm];
      float s = wv / (1.f + expf(-q));
      s += __shfl_xor(s, 1, 16);
      s += __shfl_xor(s, 2, 16);
      s += __shfl_xor(s, 4, 16);
      s += __shfl_xor(s, 8, 16);
      asum[i] += s;
    }
  }
  if (lm == 0) {
#pragma unroll
    for (int i = 0; i < 8; ++i) alpha[r0 + i + 8*half] = asum[i];
  }
}

// ---------------------------------------------------------------------------
// 10) alpha-weighted pool + concat [a, ht] -> cat[B,2H]
// ---------------------------------------------------------------------------
__global__ void k_apool(const float* __restrict__ alpha, const float* __restrict__ seqb,
                        const int* __restrict__ mask, const float* __restrict__ ht,
                        float* __restrict__ cat) {
  int b = blockIdx.x, hh = threadIdx.x;
  float acc = 0.f;
  for (int l = 0; l < LL; ++l)
    acc += alpha[b*LL + l] * seqb[((size_t)b*LL + l)*HH + hh] * (float)mask[b*LL + l];
  cat[(size_t)b*2*HH + hh] = acc;
  cat[(size_t)b*2*HH + HH + hh] = ht[b*HH + hh];
}

// ---------------------------------------------------------------------------
// 11) out = centered-L2 layer norm of (sl + sg)
// ---------------------------------------------------------------------------
__global__ void k_final(const float* __restrict__ sl, const float* __restrict__ sg,
                        float* __restrict__ out) {
  int b = blockIdx.x, hh = threadIdx.x;
  __shared__ float sh[HH];
  __shared__ float smean, sinv;
  float v = sl[b*HH + hh] + sg[b*HH + hh];
  sh[hh] = v; __syncthreads();
  for (int s = HH/2; s > 0; s >>= 1) { if (hh < s) sh[hh] += sh[hh + s]; __syncthreads(); }
  if (hh == 0) smean = sh[0] / (float)HH;
  __syncthreads();
  float c = v - smean;
  sh[hh] = c * c; __syncthreads();
  for (int s = HH/2; s > 0; s >>= 1) { if (hh < s) sh[hh] += sh[hh + s]; __syncthreads(); }
  if (hh == 0) sinv = rsqrtf(sh[0]);
  __syncthreads();
  out[b*HH + hh] = c * sinv;
}

// ---------------------------------------------------------------------------
extern "C" void kernel_launch(void* const* d_in, const int* in_sizes, int n_in,
                              void* d_out, int out_size, void* d_ws, size_t ws_size,
                              hipStream_t stream) {
  (void)in_sizes; (void)n_in; (void)out_size; (void)ws_size;
  const int*   alias_in = (const int*)d_in[0];
  const int*   items    = (const int*)d_in[1];
  const int*   mask     = (const int*)d_in[2];
  const int*   cates    = (const int*)d_in[3];
  const int*   eidx     = (const int*)d_in[4];
  const int*   etype    = (const int*)d_in[5];
  const float* eattr    = (const float*)d_in[6];
  const float* item_emb = (const float*)d_in[7];
  const float* cate_emb = (const float*)d_in[8];
  const float* pos_emb  = (const float*)d_in[9];
  const float* len_emb  = (const float*)d_in[10];
  const float* rel_emb  = (const float*)d_in[11];
  const float* se_W     = (const float*)d_in[12];  // [2,4,H,H]
  const float* se_b     = (const float*)d_in[13];  // [2,4,H]
  const float* se_w5    = (const float*)d_in[14];  // [2,H]
  const float* se_Wt    = (const float*)d_in[15];  // [2,2H,H]
  const float* se_bt    = (const float*)d_in[16];  // [2,H]
  const float* gat_W    = (const float*)d_in[17];  // [H,H]

  float* ws = (float*)d_ws;
  const size_t BLHf = (size_t)BB * LL * HH;        // 6,553,600 floats
  float*    seqb  = ws;                            // seq_hidden [B,L,H]
  float*    hbuf  = ws + BLHf;                     // GAT h [N,H]; reused as g_seq
  float*    gout  = ws + 2*BLHf;                   // graph_emb [N,H]
  float*    logit = ws + 3*BLHf;                   // [E]
  float*    exb   = logit + NEDGE;                 // [E]
  float*    den   = exb + NEDGE;                   // [N]
  unsigned* mx    = (unsigned*)(den + NNODES);     // [N]
  float*    mi    = den + 2*NNODES;                // mean_item [B,H]
  float*    mc    = mi + BB*HH;                    // mean_cate [B,H]
  float*    sn    = mc + BB*HH;                    // s_node [B,H]
  float*    htl   = sn + BB*HH;
  float*    htg   = htl + BB*HH;
  float*    qc    = htg + BB*HH;
  float*    alpha = qc + BB*HH;                    // [B*L]
  float*    cat   = alpha + BB*LL;                 // [B,2H]
  float*    sl    = cat + (size_t)BB*2*HH;
  float*    sg    = sl + BB*HH;

  dim3 b256(256), b32(32);

  // stage 1: embeddings / pooling / seq_hidden
  k_prep<<<BB, b256, 0, stream>>>(alias_in, items, mask, cates, item_emb, cate_emb,
                                  pos_emb, len_emb, seqb, mi, mc, htl);
  k_zero<<<(NNODES*HH)/256, b256, 0, stream>>>(gout, den, mx);

  // stage 2: GAT — gather-fused WMMA GEMM, then segment softmax + scatter
  k_gemm<<<dim3(NNODES/16, 16), b32, 0, stream>>>(item_emb, items, gat_W, nullptr, hbuf, HH);
  k_edge_logit<<<NEDGE/8, b256, 0, stream>>>(hbuf, eidx, etype, eattr, rel_emb, logit, mx);
  k_edge_exp<<<NEDGE/256, b256, 0, stream>>>(eidx, logit, mx, exb, den);
  k_edge_scatter<<<NEDGE, b256, 0, stream>>>(hbuf, eidx, etype, exb, den, rel_emb, gout);
  k_gpool<<<BB, b256, 0, stream>>>(alias_in, items, mask, gout, sn, hbuf, htg); // hbuf := g_seq

  // stage 3: local session encoder (enc 0)
  k_qc<<<BB, b256, 0, stream>>>(htl, mi, mc, se_W, se_b, qc);
  k_alpha<<<NNODES/16, b32, 0, stream>>>(seqb, qc, se_W + 2*HH*HH, se_w5, alpha);
  k_apool<<<BB, b256, 0, stream>>>(alpha, seqb, mask, htl, cat);
  k_gemm<<<dim3(BB/16, 16), b32, 0, stream>>>(cat, nullptr, se_Wt, se_bt, sl, 2*HH);

  // stage 4: global session encoder (enc 1) — mean_cate := s_node
  k_qc<<<BB, b256, 0, stream>>>(htg, mi, sn, se_W + 4*HH*HH, se_b + 4*HH, qc);
  k_alpha<<<NNODES/16, b32, 0, stream>>>(hbuf, qc, se_W + 6*HH*HH, se_w5 + HH, alpha);
  k_apool<<<BB, b256, 0, stream>>>(alpha, hbuf, mask, htg, cat);
  k_gemm<<<dim3(BB/16, 16), b32, 0, stream>>>(cat, nullptr, se_Wt + 2*HH*HH, se_bt + HH, sg, 2*HH);

  // stage 5: layer norm
  k_final<<<BB, b256, 0, stream>>>(sl, sg, (float*)d_out);
}